// grid_layer_21457656610895
// MI455X (gfx1250) — compile-verified
//
#include <hip/hip_runtime.h>

typedef __attribute__((ext_vector_type(2))) float v2f;
typedef __attribute__((ext_vector_type(8))) float v8f;

#define N_NODES 65536
#define NHN 7
#define NBATCH 4
#define NVE 64                    // NV*E contiguous floats per (b, node)
#define EMB 32
#define PAIRS (N_NODES * NHN)     // 458752
#define TILES (PAIRS / 16)        // 28672
#define WPB 8                     // waves per block
#define BLOCKS (TILES / WPB)      // 3584

// sigmoid with raw hardware v_rcp_f32 (1 ulp) instead of IEEE divide chain
__device__ __forceinline__ float sigmoidf_fast(float v) {
    return __builtin_amdgcn_rcpf(1.0f + __expf(-v));
}

__global__ __launch_bounds__(256)
void grid_layer_fused_kernel(const float* __restrict__ x,
                             const float* __restrict__ coords,
                             const float* __restrict__ w1,
                             const float* __restrict__ b1,
                             const float* __restrict__ w2,
                             const int*   __restrict__ adjc,
                             const unsigned char* __restrict__ valid,
                             float* __restrict__ out)
{
    __shared__ float s_dx[WPB][16];
    __shared__ float s_dy[WPB][16];
    __shared__ float s_sc[WPB][16];
    __shared__ int   s_ix[WPB][16];
    __shared__ float s_emb[WPB][16][EMB];

    const int lane = threadIdx.x & 31;
    const int wv   = threadIdx.x >> 5;
    const int tile = blockIdx.x * WPB + wv;
    const int pbase = tile * 16;

    const int m  = lane & 15;    // row within 16x16 tile / column index
    const int hi = lane >> 4;    // upper half-wave selector

    // ---- Phase 1: gather coords / indices for 16 (n,h) pairs (lanes 0..15)
    if (lane < 16) {
        int p   = pbase + lane;          // flat (n*NH + h); adjc/valid contiguous in p
        int n   = p / NHN;
        int idx = adjc[p];
        float cxn = coords[n];
        float cyn = coords[N_NODES + n];
        float dx  = coords[idx]           - cxn;   // adjc[:,0] == n -> center = coords[n]
        float dy  = coords[N_NODES + idx] - cyn;
        if (fabsf(dx) < 1e-6f) dx = 0.0f;
        if (fabsf(dy) < 1e-6f) dy = 0.0f;
        s_dx[wv][lane] = dx;
        s_dy[wv][lane] = dy;
        s_ix[wv][lane] = idx;
        s_sc[wv][lane] = valid[p] ? 1.0f : 0.0f;
    }
    __builtin_amdgcn_wave_barrier();   // per-wave LDS regions; LDS in-order within wave

    // ---- Phase 2: H = silu(P @ w1 + b1), then emb_pre = H @ w2 via fp32 WMMA
    // A fragment (16x4 f32): lane holds row m, K = 4*kk + 2*hi + {0,1}
    const float dxm = s_dx[wv][m];
    const float dym = s_dy[wv][m];

    v8f acc0 = {};   // N cols 0..15
    v8f acc1 = {};   // N cols 16..31

#pragma unroll
    for (int kk = 0; kk < 8; ++kk) {
        const int k0 = kk * 4 + hi * 2;
        float pre0 = dxm * w1[k0]     + dym * w1[EMB + k0]     + b1[k0];
        float pre1 = dxm * w1[k0 + 1] + dym * w1[EMB + k0 + 1] + b1[k0 + 1];
        v2f a;
        a.x = pre0 * sigmoidf_fast(pre0);   // silu
        a.y = pre1 * sigmoidf_fast(pre1);

        v2f bt0, bt1;   // B fragment: rows k0,k0+1 of w2, cols m / m+16
        bt0.x = w2[k0 * EMB + m];
        bt0.y = w2[(k0 + 1) * EMB + m];
        bt1.x = w2[k0 * EMB + 16 + m];
        bt1.y = w2[(k0 + 1) * EMB + 16 + m];

        acc0 = __builtin_amdgcn_wmma_f32_16x16x4_f32(false, a, false, bt0,
                                                     (short)0, acc0, false, false);
        acc1 = __builtin_amdgcn_wmma_f32_16x16x4_f32(false, a, false, bt1,
                                                     (short)0, acc1, false, false);
    }

    // C layout: VGPR i -> row (i + 8*hi), col m (tile0) / m+16 (tile1)
#pragma unroll
    for (int i = 0; i < 8; ++i) {
        const int row = i + 8 * hi;
        s_emb[wv][row][m]      = 16.0f * sigmoidf_fast(acc0[i]);
        s_emb[wv][row][m + 16] = 16.0f * sigmoidf_fast(acc1[i]);
    }
    __builtin_amdgcn_wave_barrier();

    // ---- Phase 3: out[b, p, v, e] = x[b, idx, v, e] * emb[p, e] * mask[p]
    // lane owns contiguous elements {2*lane, 2*lane+1} of the 64-float (NV,E) block
    const int j  = lane * 2;        // 0..62 within (v,e) block
    const int ee = j & 31;          // embedding column (emb depends on e only)

#pragma unroll 2
    for (int r = 0; r < 16; ++r) {
        const int   p   = pbase + r;
        const int   idx = s_ix[wv][r];
        const float sc  = s_sc[wv][r];
        v2f ev;
        ev.x = s_emb[wv][r][ee]     * sc;
        ev.y = s_emb[wv][r][ee + 1] * sc;

        const float* xp = x   + (size_t)idx * NVE + j;
        float*       op = out + (size_t)p   * NVE + j;

#pragma unroll
        for (int b = 0; b < NBATCH; ++b) {
            // x: regular-temporal (64 MB with ~7x reuse -> keep in 192 MB L2)
            v2f xv = *reinterpret_cast<const v2f*>(xp + (size_t)b * ((size_t)N_NODES * NVE));
            v2f ov = xv * ev;
            // out: write-once stream (470 MB) -> non-temporal, don't evict x from L2
            __builtin_nontemporal_store(ov,
                reinterpret_cast<v2f*>(op + (size_t)b * ((size_t)PAIRS * NVE)));
        }
    }
}

extern "C" void kernel_launch(void* const* d_in, const int* in_sizes, int n_in,
                              void* d_out, int out_size, void* d_ws, size_t ws_size,
                              hipStream_t stream) {
    (void)in_sizes; (void)n_in; (void)out_size; (void)d_ws; (void)ws_size;
    const float*         x      = (const float*)d_in[0];
    const float*         coords = (const float*)d_in[1];
    const float*         w1     = (const float*)d_in[2];
    const float*         b1     = (const float*)d_in[3];
    const float*         w2     = (const float*)d_in[4];
    const int*           adjc   = (const int*)d_in[5];
    const unsigned char* valid  = (const unsigned char*)d_in[6];
    float*               out    = (float*)d_out;

    grid_layer_fused_kernel<<<BLOCKS, 256, 0, stream>>>(x, coords, w1, b1, w2,
                                                        adjc, valid, out);
}